// MultiHeadAttention_38225208935076
// MI455X (gfx1250) — compile-verified
//
#include <hip/hip_runtime.h>

typedef float v2f __attribute__((ext_vector_type(2)));
typedef float v8f __attribute__((ext_vector_type(8)));

#define B_  16
#define N_  1024
#define E_  768
#define E3_ 2304   // 3*E, row stride of Wqkv

// ---------------------------------------------------------------------------
// Stage 1: partial[b*16 + chunk][e] = sum over 64 tokens of x[b, n, e]
// 256 blocks (b x 16 chunks), 256 threads; coalesced 128B/wave loads.
// ---------------------------------------------------------------------------
__global__ __launch_bounds__(256) void reduce_stage1(const float* __restrict__ x,
                                                     float* __restrict__ partial) {
    const int blk = blockIdx.x;        // 0..255
    const int b   = blk >> 4;
    const int ch  = blk & 15;
    const int tid = threadIdx.x;       // 0..255

    float acc0 = 0.f, acc1 = 0.f, acc2 = 0.f;
    const float* xp = x + ((size_t)b * N_ + (size_t)ch * 64) * E_;
    for (int n = 0; n < 64; ++n) {
        const float* row = xp + (size_t)n * E_;
        __builtin_prefetch(row + E_, 0, 1);   // global_prefetch next token row
        acc0 += row[tid];
        acc1 += row[256 + tid];
        acc2 += row[512 + tid];
    }
    float* pp = partial + (size_t)blk * E_;
    pp[tid]       = acc0;
    pp[256 + tid] = acc1;
    pp[512 + tid] = acc2;
}

// ---------------------------------------------------------------------------
// Stage 2: xsum[b][e] = sum over 16 chunks of partial[b*16+ch][e]
// ---------------------------------------------------------------------------
__global__ __launch_bounds__(256) void reduce_stage2(const float* __restrict__ partial,
                                                     float* __restrict__ xsum) {
    const int idx = blockIdx.x * 256 + threadIdx.x;   // 0..12287
    const int b   = idx / E_;
    const int e   = idx - b * E_;
    const float* pp = partial + (size_t)b * 16 * E_ + e;
    float acc = 0.f;
#pragma unroll
    for (int ch = 0; ch < 16; ++ch) acc += pp[(size_t)ch * E_];
    xsum[idx] = acc;
}

// ---------------------------------------------------------------------------
// D(16 x 768) = A(16 x 768, lda=768) @ W(768 x 768 slice, row stride ldw)
//               [+ bias broadcast over rows]
// One wave per 16-column tile; V_WMMA_F32_16X16X4_F32, K stepped by 4.
// fp32 A layout : lane l -> M = l&15,  VGPR j -> K = 2*(l>>4)+j
// fp32 B layout : lane l -> N = l&15,  VGPR j -> K = 2*(l>>4)+j
// fp32 C/D      : lane l -> N = l&15,  VGPR j -> M = j + 8*(l>>4)
// ---------------------------------------------------------------------------
__global__ __launch_bounds__(32) void gemm16_wmma(const float* __restrict__ A,
                                                  const float* __restrict__ W,
                                                  int ldw,
                                                  const float* __restrict__ bias,
                                                  float* __restrict__ D) {
    const int col0 = blockIdx.x * 16;       // 48 tiles over 768 columns
    const int l    = threadIdx.x;           // 0..31
    const int m    = l & 15;
    const int g    = l >> 4;                // lane group (0/1)
    const int n    = col0 + (l & 15);

    const float binit = bias ? bias[n] : 0.0f;   // uniform branch, EXEC stays full
    v8f c;
#pragma unroll
    for (int j = 0; j < 8; ++j) c[j] = binit;

    for (int k0 = 0; k0 < E_; k0 += 4) {
        v2f a, bb;
        const float* ap = A + (size_t)m * E_ + k0 + 2 * g;
        a[0]  = ap[0];
        a[1]  = ap[1];
        bb[0] = W[(size_t)(k0 + 2 * g)     * ldw + n];
        bb[1] = W[(size_t)(k0 + 2 * g + 1) * ldw + n];
        c = __builtin_amdgcn_wmma_f32_16x16x4_f32(false, a, false, bb,
                                                  (short)0, c, false, false);
    }

#pragma unroll
    for (int j = 0; j < 8; ++j)
        D[(size_t)(j + 8 * g) * E_ + n] = c[j];
}

// ---------------------------------------------------------------------------
// out[b, n, :] = Y[b, :] for all n. float4 stores, 3KB contiguous per row.
// Grid: B * (N/16) blocks, 192 threads (768 floats = 192 float4 per row).
// ---------------------------------------------------------------------------
__global__ __launch_bounds__(192) void broadcast_out(const float* __restrict__ Y,
                                                     float* __restrict__ out) {
    const int blk = blockIdx.x;          // 0..1023
    const int b   = blk >> 6;            // / (N/16 = 64)
    const int nc  = blk & 63;
    const int t   = threadIdx.x;         // 0..191

    const float4 v = ((const float4*)(Y + (size_t)b * E_))[t];
    float4* o4 = (float4*)(out + ((size_t)b * N_ + (size_t)nc * 16) * E_);
#pragma unroll
    for (int n = 0; n < 16; ++n)
        o4[(size_t)n * (E_ / 4) + t] = v;
}

// ---------------------------------------------------------------------------
extern "C" void kernel_launch(void* const* d_in, const int* in_sizes, int n_in,
                              void* d_out, int out_size, void* d_ws, size_t ws_size,
                              hipStream_t stream) {
    const float* x    = (const float*)d_in[0];   // [B, N, E]
    const float* Wqkv = (const float*)d_in[1];   // [E, 3E] row-major
    const float* Wo   = (const float*)d_in[2];   // [E, E]
    const float* bo   = (const float*)d_in[3];   // [E]
    float*       out  = (float*)d_out;           // [B, N, E]

    float* ws      = (float*)d_ws;
    float* partial = ws;                          // 16*16*768 = 196608 floats
    float* xsum    = partial + B_ * 16 * E_;      // 16*768
    float* S       = xsum    + B_ * E_;           // 16*768
    float* Y       = S       + B_ * E_;           // 16*768

    const float* Wv = Wqkv + 2 * E_;              // V column-slice, row stride 3E

    // 1) xsum[b] = sum_n x[b,n,:]   (two-stage, no atomics)
    reduce_stage1<<<B_ * 16, 256, 0, stream>>>(x, partial);
    reduce_stage2<<<(B_ * E_) / 256, 256, 0, stream>>>(partial, xsum);

    // 2) S = xsum @ Wv              (16x768 @ 768x768, WMMA fp32)
    gemm16_wmma<<<E_ / 16, 32, 0, stream>>>(xsum, Wv, E3_, nullptr, S);

    // 3) Y = S @ Wo + bo            (16x768 @ 768x768, WMMA fp32)
    gemm16_wmma<<<E_ / 16, 32, 0, stream>>>(S, Wo, E_, bo, Y);

    // 4) out[b,n,:] = Y[b,:]        (50 MB broadcast store)
    broadcast_out<<<B_ * (N_ / 16), 192, 0, stream>>>(Y, out);
}